// History_1786706395394
// MI455X (gfx1250) — compile-verified
//
#include <hip/hip_runtime.h>

// ---------------------------------------------------------------------------
// Segmented mean (counts <= 16 per group) + first-row gather, mapped to
// V_WMMA_F32_16X16X4_F32: means = W(16xK) x X(Kx128), W = selection matrix
// holding 1/count.  fp32 end-to-end; bandwidth-bound (~705 MB total ->
// ~30us floor @ 23.3 TB/s).  Hot loop is branch-free: weights computed by
// per-lane range compares (no LDS in loop), dual accumulators, unconditional
// speculative prefetch.
// ---------------------------------------------------------------------------

typedef float v2f __attribute__((ext_vector_type(2)));
typedef float v8f __attribute__((ext_vector_type(8)));

#define D_LOC 128
#define D_TIM 64
#define D_OUT (D_LOC + D_TIM)   // 192
#define GPB   16                // groups per block (one WMMA M-tile)

// --- kernel 1: per-16-group block sums of counts ---------------------------
__global__ void k_blocksums(const int* __restrict__ cnt, int* __restrict__ bsum,
                            int nblocks) {
    int b = blockIdx.x * blockDim.x + threadIdx.x;
    if (b < nblocks) {
        const int* p = cnt + b * GPB;
        int s = 0;
#pragma unroll
        for (int i = 0; i < GPB; ++i) s += p[i];
        bsum[b] = s;
    }
}

// --- kernel 2: exclusive scan of block sums (single 1024-thread WG) --------
__global__ void __launch_bounds__(1024) k_scan(const int* __restrict__ bsum,
                                               int* __restrict__ bstart, int n) {
    __shared__ int part[1024];
    const int t = threadIdx.x;
    const int per = (n + 1023) >> 10;
    const int base = t * per;
    int s = 0;
    for (int i = 0; i < per; ++i) {
        int idx = base + i;
        s += (idx < n) ? bsum[idx] : 0;
    }
    part[t] = s;
    __syncthreads();
    for (int off = 1; off < 1024; off <<= 1) {
        int v = part[t];
        int add = (t >= off) ? part[t - off] : 0;
        __syncthreads();
        part[t] = v + add;
        __syncthreads();
    }
    int ex = (t > 0) ? part[t - 1] : 0;
    for (int i = 0; i < per; ++i) {
        int idx = base + i;
        if (idx < n) { bstart[idx] = ex; ex += bsum[idx]; }
    }
}

// --- kernel 3: main WMMA segmented-mean kernel -----------------------------
// One workgroup = 8 wave32 = 16 groups.  Wave w owns output columns
// [16w, 16w+16).  Branch-free main loop over full 4-row quads; guarded tail.
__global__ void __launch_bounds__(256) k_main(const float* __restrict__ loc,
                                              const float* __restrict__ tim,
                                              const int*   __restrict__ cnt,
                                              const int*   __restrict__ bstart,
                                              float*       __restrict__ out) {
    const int blk   = blockIdx.x;
    const int gbase = blk * GPB;
    const int t     = threadIdx.x;

    __shared__ int   s_start[GPB + 1];   // exclusive local row starts
    __shared__ float s_invc[GPB];        // 1/count per group

    if (t == 0) {
        int acc = 0;
#pragma unroll
        for (int g = 0; g < GPB; ++g) { s_start[g] = acc; acc += cnt[gbase + g]; }
        s_start[GPB] = acc;
    }
    __syncthreads();
    if (t < GPB) s_invc[t] = 1.0f / (float)(s_start[t + 1] - s_start[t]);
    __syncthreads();

    const int rowStart = bstart[blk];    // first global row of this block
    const int rows     = s_start[GPB];   // rows covered by these 16 groups
    const int lane     = t & 31;
    const int wave     = t >> 5;         // column tile 0..7
    const int m        = lane & 15;      // group index (A row / D column pos)
    const int koff     = (lane >> 4) * 2;// K-pair selector per A/B layout
    const int col      = wave * 16 + m;  // D_LOC column this lane produces

    // Loop-invariant per-lane weight parameters: row k belongs to group m
    // iff (unsigned)(k - sreg) < cm; weight is then invm.
    const int      sreg = s_start[m];
    const unsigned cm   = (unsigned)(s_start[m + 1] - sreg);
    const float    invm = s_invc[m];

    const float* locBase = loc + (size_t)rowStart * D_LOC;

    v8f acc0 = {};
    v8f acc1 = {};
    const int q = rows >> 2;             // full quad steps (no bounds checks)
#pragma unroll 2
    for (int st = 0; st < q; ++st) {
        const int k0 = st * 4 + koff;
        const int k1 = k0 + 1;
        const float b0 = locBase[(size_t)k0 * D_LOC + col];
        const float b1 = locBase[(size_t)k1 * D_LOC + col];
        const float a0 = ((unsigned)(k0 - sreg) < cm) ? invm : 0.f;
        const float a1 = ((unsigned)(k1 - sreg) < cm) ? invm : 0.f;
        __builtin_prefetch(&locBase[(size_t)(k0 + 16) * D_LOC + col], 0, 3);

        v2f A; A.x = a0; A.y = a1;       // A 16x4: lane half selects K pair
        v2f B; B.x = b0; B.y = b1;       // B 4x16: same K pairing, N = lane%16
        if (st & 1) {
            acc1 = __builtin_amdgcn_wmma_f32_16x16x4_f32(
                false, A, false, B, (short)0, acc1, false, false);
        } else {
            acc0 = __builtin_amdgcn_wmma_f32_16x16x4_f32(
                false, A, false, B, (short)0, acc0, false, false);
        }
    }

    // Tail: rows % 4 leftover rows (guarded; exec restored before WMMA).
    const int rem = rows - (q << 2);
    if (rem > 0) {
        const int k0 = (q << 2) + koff;
        const int k1 = k0 + 1;
        float a0 = 0.f, a1 = 0.f, b0 = 0.f, b1 = 0.f;
        if (koff < rem) {
            a0 = ((unsigned)(k0 - sreg) < cm) ? invm : 0.f;
            b0 = locBase[(size_t)k0 * D_LOC + col];
        }
        if (koff + 1 < rem) {
            a1 = ((unsigned)(k1 - sreg) < cm) ? invm : 0.f;
            b1 = locBase[(size_t)k1 * D_LOC + col];
        }
        v2f A; A.x = a0; A.y = a1;
        v2f B; B.x = b0; B.y = b1;
        acc0 = __builtin_amdgcn_wmma_f32_16x16x4_f32(
            false, A, false, B, (short)0, acc0, false, false);
    }

    const v8f total = acc0 + acc1;

    // D layout: VGPR v, lane l -> group = v + 8*(l>=16), N = l&15.
#pragma unroll
    for (int v = 0; v < 8; ++v) {
        int g = (lane < 16) ? v : (v + 8);
        out[(size_t)(gbase + g) * D_OUT + col] = total[v];
    }

    // tim_first: 16 threads per group copy its 64-col first row (float4).
    {
        int g    = t >> 4;
        int c4   = (t & 15) * 4;
        int grow = rowStart + s_start[g];
        const float4* src = (const float4*)(tim + (size_t)grow * D_TIM + c4);
        float4*       dst = (float4*)(out + (size_t)(gbase + g) * D_OUT + D_LOC + c4);
        *dst = *src;
    }
}

// ---------------------------------------------------------------------------
extern "C" void kernel_launch(void* const* d_in, const int* in_sizes, int n_in,
                              void* d_out, int out_size, void* d_ws, size_t ws_size,
                              hipStream_t stream) {
    const float* loc = (const float*)d_in[0];
    const float* tim = (const float*)d_in[1];
    const int*   cnt = (const int*)d_in[2];
    float*       out = (float*)d_out;

    const int ngroups = in_sizes[2];           // history_count elements
    const int nblocks = ngroups / GPB;         // 8192 for the reference shapes

    int* bsum   = (int*)d_ws;                  // nblocks ints
    int* bstart = bsum + nblocks;              // nblocks ints (64 KB total)

    k_blocksums<<<(nblocks + 255) / 256, 256, 0, stream>>>(cnt, bsum, nblocks);
    k_scan<<<1, 1024, 0, stream>>>(bsum, bstart, nblocks);
    k_main<<<nblocks, 256, 0, stream>>>(loc, tim, cnt, bstart, out);
}